// LinearSplineLayer_5583457485588
// MI455X (gfx1250) — compile-verified
//
#include <hip/hip_runtime.h>
#include <cstdint>

// LinearSplineLayer forward for MI455X (gfx1250, wave32).
// B=32768 rows, D=64 dims, G=64 uniform bins.
// Memory-bound: ~16MB total traffic -> ~0.7us at 23.3 TB/s. Design:
//  1) tiny kernel builds per-dim (cdf, slope) tables (softmax+cumsum) in d_ws
//  2) main kernel TDM-loads the 33KB table into LDS (tensor_load_to_lds),
//     streams x with float2/wave-per-row coalesced loads, gathers from LDS,
//     reduces logdet with wave32 shuffles.

#define D_DIM 64
#define G_BINS 64
#define TAB_ENTRIES (D_DIM * (G_BINS + 1)) // 4160 float2 = 33280 bytes
#define INV_H (1.0f / (1.0f / (float)G_BINS + 1e-7f))

typedef unsigned int v4u __attribute__((ext_vector_type(4)));
typedef int          v8i __attribute__((ext_vector_type(8)));
typedef int          v4i __attribute__((ext_vector_type(4)));

// ---------------------------------------------------------------------------
// Kernel 1: per-dim softmax + cumsum -> table[d][g] = {cdf_g, slope_g},
// table[d][G] = {cdf_G (=1), 1.0} for the right tail. 64 threads, trivial.
// ---------------------------------------------------------------------------
__global__ void spline_build_tables(const float* __restrict__ pdf,
                                    float2* __restrict__ tab) {
    int d = threadIdx.x;
    if (d >= D_DIM) return;
    const float* row = pdf + d * G_BINS;

    float m = -3.402823466e+38f;
    for (int g = 0; g < G_BINS; ++g) m = fmaxf(m, row[g]);
    float sum = 0.0f;
    for (int g = 0; g < G_BINS; ++g) sum += __expf(row[g] - m);
    float inv = 1.0f / sum;

    float cdf = 0.0f;
    for (int g = 0; g < G_BINS; ++g) {
        float p = __expf(row[g] - m) * inv;
        tab[d * (G_BINS + 1) + g] = make_float2(cdf, p * INV_H);
        cdf += p;
    }
    tab[d * (G_BINS + 1) + G_BINS] = make_float2(cdf, 1.0f);
}

// ---------------------------------------------------------------------------
// Kernel 2: main stream. One wave32 per row (64 dims -> float2 per lane).
// Table staged into LDS via Tensor Data Mover (CDNA5 async-tensor path).
// ---------------------------------------------------------------------------
__launch_bounds__(1024, 1)
__global__ void spline_fwd(const float* __restrict__ x,
                           const float2* __restrict__ tab_g,
                           float* __restrict__ y,
                           float* __restrict__ logdet,
                           int B) {
    __shared__ float2 tab[TAB_ENTRIES];

    // --- TDM stage: one wave issues a single 33KB tensor_load_to_lds ------
    if (threadIdx.x < 32u) {
        uint32_t lds_off = (uint32_t)(uintptr_t)(void*)&tab[0]; // flat addr[31:0] == LDS offset
        uint64_t ga      = (uint64_t)(uintptr_t)tab_g;

        // D# group 0 (128b): count=1 | lds_addr | global_addr[56:0] | type=2
        v4u g0;
        g0[0] = 1u;                                            // count=1, user descriptor
        g0[1] = lds_off;                                       // lds_addr (bytes)
        g0[2] = (uint32_t)(ga & 0xffffffffu);                  // global_addr[31:0] -> bits 95:64
        g0[3] = (uint32_t)((ga >> 32) & 0x01ffffffu)           // global_addr[56:32] -> bits 120:96
              | (2u << 30);                                    // type=2 ("image")

        // D# group 1 (256b): 1-D copy, data_size=8B, dims/strides = 4160 elems
        v8i g1;
        g1[0] = (3 << 16);                                     // data_size=3 (8 bytes), wg_mask=0
        g1[1] = (int)((TAB_ENTRIES & 0xffffu) << 16);          // tensor_dim0[15:0] @ bits 63:48
        g1[2] = (int)(((TAB_ENTRIES >> 16) & 0xffffu))         // tensor_dim0[31:16]
              | (1 << 16);                                     // tensor_dim1 = 1
        g1[3] = (int)((TAB_ENTRIES & 0xffffu) << 16);          // tile_dim0 = 4160 @ bits 127:112
        g1[4] = 1;                                             // tile_dim1 = 1, tile_dim2 = 0
        g1[5] = TAB_ENTRIES;                                   // tensor_dim0_stride[31:0]
        g1[6] = (int)((TAB_ENTRIES & 0xffffu) << 16);          // dim0_stride[47:32]=0, dim1_stride[15:0]
        g1[7] = 0;                                             // dim1_stride[47:16]=0

        v4i z4 = {0, 0, 0, 0};                                 // groups 2/3 unused (<=2D tensor)
        v8i z8 = {0, 0, 0, 0, 0, 0, 0, 0};                     // extra group (clang-23 6-arg form)
        __builtin_amdgcn_tensor_load_to_lds(g0, g1, z4, z4, z8, 0);
        __builtin_amdgcn_s_wait_tensorcnt(0);                  // s_wait_tensorcnt 0
    }
    __syncthreads();

    const int lane  = (int)(threadIdx.x & 31u);
    const int wave  = (int)(threadIdx.x >> 5);
    const int wpb   = (int)(blockDim.x >> 5);
    const int gwave = (int)blockIdx.x * wpb + wave;
    const int nwav  = (int)gridDim.x * wpb;

    for (int row = gwave; row < B; row += nwav) {
        // coalesced: wave reads 32 x float2 = one 256B row of x
        float2 xv = ((const float2*)(x + (size_t)row * D_DIM))[lane];

        int nrow = row + nwav;
        if (nrow < B)
            __builtin_prefetch(x + (size_t)nrow * D_DIM + 2 * lane, 0, 0); // global_prefetch_b8

        float xe[2] = {xv.x, xv.y};
        float yo[2];
        float acc = 0.0f;

#pragma unroll
        for (int e = 0; e < 2; ++e) {
            const int d  = 2 * lane + e;
            float xs = (xe[e] + 1.0f) * 0.5f;           // rescale to [0,1]
            bool left  = xs < 0.0f;
            bool right = xs >= 1.0f;
            int idx = (int)floorf(xs * (float)G_BINS);  // exact: G power of two
            idx = idx < 0 ? 0 : (idx > G_BINS - 1 ? G_BINS - 1 : idx);

            float2 cs = tab[d * (G_BINS + 1) + idx];    // {cdf_i, slope_i}
            float2 ct = tab[d * (G_BINS + 1) + G_BINS]; // {cdf_G, 1}

            float yu;
            if (left)       yu = xs;                                   // slope 1 tail
            else if (right) yu = ct.x + (xs - 1.0f);                   // slope 1 tail
            else            yu = cs.x + (xs - (float)idx * (1.0f / G_BINS)) * cs.y;

            acc   += (left || right) ? 0.0f : __logf(cs.y);
            yo[e]  = yu * 2.0f - 1.0f;                  // un-rescale
        }

        ((float2*)(y + (size_t)row * D_DIM))[lane] = make_float2(yo[0], yo[1]);

        // wave32 row reduction for logdet[row]
#pragma unroll
        for (int off = 16; off > 0; off >>= 1)
            acc += __shfl_xor(acc, off, 32);
        if (lane == 0) logdet[row] = acc;
    }
}

// ---------------------------------------------------------------------------
extern "C" void kernel_launch(void* const* d_in, const int* in_sizes, int n_in,
                              void* d_out, int out_size, void* d_ws, size_t ws_size,
                              hipStream_t stream) {
    const float* x   = (const float*)d_in[0];   // [B, 64]
    const float* pdf = (const float*)d_in[1];   // [1, 64, 64]
    const int B = in_sizes[0] / D_DIM;

    float2* tab   = (float2*)d_ws;              // 33,280 bytes
    float* y      = (float*)d_out;              // first B*64 floats
    float* logdet = y + (size_t)B * D_DIM;      // then B floats

    spline_build_tables<<<1, D_DIM, 0, stream>>>(pdf, tab);
    spline_fwd<<<256, 1024, 0, stream>>>(x, tab, y, logdet, B);
}